// SelectiveSSM_43679817401114
// MI455X (gfx1250) — compile-verified
//
#include <hip/hip_runtime.h>
#include <hip/hip_bf16.h>
#include <math.h>

// ---------------- problem constants ----------------
#define BATCH   2
#define LENGTH  2048
#define DINNER  1024
#define DSTATE  16
#define DTRANK  64
#define MROWS   (BATCH * LENGTH)      // 4096
#define XPROJ_N (DTRANK + 2 * DSTATE) // 96

// chunked scan
#define NCH 16
#define LCH (LENGTH / NCH)            // 128

typedef __attribute__((ext_vector_type(2))) float v2f;
typedef __attribute__((ext_vector_type(8))) float v8f;

// ---------------------------------------------------------------------------
// Kernel 1: x_dbl[m][j] = sum_k u[m][k] * W_x[j][k]   (M=4096, N=96, K=1024)
// One wave32 per 16x16 tile, V_WMMA_F32_16X16X4_F32, K-loop of 256 WMMAs.
// A layout (16x4 f32): lane l holds M=l&15, K = 2*(l>>4)+v  -> float2 load.
// B layout (4x16 f32): lane l holds N=l&15, K = 2*(l>>4)+v  (B[k][n]=W_x[n][k]).
// ---------------------------------------------------------------------------
__global__ void k_xproj(const float* __restrict__ u, const float* __restrict__ Wx,
                        float* __restrict__ xdbl) {
  const int wave = blockIdx.x * (blockDim.x >> 5) + (threadIdx.x >> 5);
  const int lane = threadIdx.x & 31;
  const int NT = XPROJ_N / 16;                  // 6 tiles in N
  const int tileM = wave / NT;
  const int tileN = wave % NT;
  if (tileM >= MROWS / 16) return;

  const int khalf = (lane >> 4) << 1;           // 0 or 2
  const float* aRow = u  + (size_t)(tileM * 16 + (lane & 15)) * DINNER + khalf;
  const float* bRow = Wx + (size_t)(tileN * 16 + (lane & 15)) * DINNER + khalf;

  v8f acc = {};
  for (int k = 0; k < DINNER; k += 4) {
    v2f a = *(const v2f*)(aRow + k);
    v2f b = *(const v2f*)(bRow + k);
    acc = __builtin_amdgcn_wmma_f32_16x16x4_f32(
        /*neg_a=*/false, a, /*neg_b=*/false, b,
        /*c_mod=*/(short)0, acc, /*reuse_a=*/false, /*reuse_b=*/false);
  }
  // C/D layout: vgpr r, lanes 0-15 -> M=r, lanes 16-31 -> M=8+r; N = lane&15
  const int col   = tileN * 16 + (lane & 15);
  const int rbase = tileM * 16 + ((lane >> 4) << 3);
#pragma unroll
  for (int r = 0; r < 8; ++r)
    xdbl[(size_t)(rbase + r) * XPROJ_N + col] = acc[r];
}

// ---------------------------------------------------------------------------
// Kernel 2: delta = softplus(delta_r @ W_dt.T + b_dt)  (M=4096, N=1024, K=64)
// delta_r = first 64 columns of x_dbl (row stride 96).
// ---------------------------------------------------------------------------
__global__ void k_dtproj(const float* __restrict__ xdbl, const float* __restrict__ Wdt,
                         const float* __restrict__ bdt, float* __restrict__ delta) {
  const int wave = blockIdx.x * (blockDim.x >> 5) + (threadIdx.x >> 5);
  const int lane = threadIdx.x & 31;
  const int NT = DINNER / 16;                   // 64 tiles in N
  const int tileM = wave / NT;
  const int tileN = wave % NT;
  if (tileM >= MROWS / 16) return;

  const int khalf = (lane >> 4) << 1;
  const float* aRow = xdbl + (size_t)(tileM * 16 + (lane & 15)) * XPROJ_N + khalf;
  const float* bRow = Wdt  + (size_t)(tileN * 16 + (lane & 15)) * DTRANK + khalf;

  v8f acc = {};
#pragma unroll
  for (int k = 0; k < DTRANK; k += 4) {
    v2f a = *(const v2f*)(aRow + k);
    v2f b = *(const v2f*)(bRow + k);
    acc = __builtin_amdgcn_wmma_f32_16x16x4_f32(
        false, a, false, b, (short)0, acc, false, false);
  }
  const int col   = tileN * 16 + (lane & 15);
  const int rbase = tileM * 16 + ((lane >> 4) << 3);
  const float bias = bdt[col];
#pragma unroll
  for (int r = 0; r < 8; ++r) {
    float x  = acc[r] + bias;
    float sp = (x > 20.f) ? x : log1pf(__expf(x));   // softplus, stable
    delta[(size_t)(rbase + r) * DINNER + col] = sp;
  }
}

// ---------------------------------------------------------------------------
// Kernel 3 (scan phase 1): per (b, d, chunk) compute prod(A_bar) and local h_end
// for each of the 16 states. One lane per d; B rows staged in LDS per chunk.
// blockIdx.x = b*64 + ch*4 + dblk   (256 threads -> 256 d-channels)
// ---------------------------------------------------------------------------
__global__ void k_scan_phase1(const float* __restrict__ u, const float* __restrict__ xdbl,
                              const float* __restrict__ delta, const float* __restrict__ logA,
                              float* __restrict__ chunkA, float* __restrict__ chunkH) {
  const int dblk = blockIdx.x & 3;
  const int ch   = (blockIdx.x >> 2) & (NCH - 1);
  const int b    = blockIdx.x >> 6;
  const int d    = dblk * 256 + threadIdx.x;
  const int t0   = ch * LCH;

  __shared__ float Bs[LCH][DSTATE];             // 8 KB
  for (int i = threadIdx.x; i < LCH * DSTATE; i += 256) {
    int t = i >> 4, n = i & 15;
    Bs[t][n] = xdbl[(size_t)(b * LENGTH + t0 + t) * XPROJ_N + DTRANK + n];
  }
  __syncthreads();

  float An[DSTATE], h[DSTATE], ap[DSTATE];
#pragma unroll
  for (int n = 0; n < DSTATE; ++n) {
    An[n] = -__expf(logA[d * DSTATE + n]);
    h[n]  = 0.f;
    ap[n] = 1.f;
  }
  for (int t = 0; t < LCH; ++t) {
    size_t idx = (size_t)(b * LENGTH + t0 + t) * DINNER + d;   // coalesced in d
    float dlt = delta[idx];
    float uu  = u[idx];
    float du  = dlt * uu;
#pragma unroll
    for (int n = 0; n < DSTATE; ++n) {
      float a = __expf(dlt * An[n]);
      h[n]  = a * h[n] + du * Bs[t][n];
      ap[n] *= a;
    }
  }
  size_t base = (((size_t)(b * DINNER + d)) * NCH + ch) * DSTATE;
#pragma unroll
  for (int n = 0; n < DSTATE; ++n) {
    chunkA[base + n] = ap[n];
    chunkH[base + n] = h[n];
  }
}

// ---------------------------------------------------------------------------
// Kernel 4 (scan phase 2): exclusive carry scan over the 16 chunks,
// one lane per (b, d, n): 32768 lanes, 16 sequential steps.
// ---------------------------------------------------------------------------
__global__ void k_scan_phase2(const float* __restrict__ chunkA, const float* __restrict__ chunkH,
                              float* __restrict__ carry) {
  const int idx = blockIdx.x * 256 + threadIdx.x;   // b*D*16 + d*16 + n
  const int n   = idx & 15;
  const int bd  = idx >> 4;
  size_t base = (size_t)bd * NCH * DSTATE + n;
  float h = 0.f;
#pragma unroll
  for (int c = 0; c < NCH; ++c) {
    carry[base + (size_t)c * DSTATE] = h;
    h = chunkA[base + (size_t)c * DSTATE] * h + chunkH[base + (size_t)c * DSTATE];
  }
}

// ---------------------------------------------------------------------------
// Kernel 5 (scan phase 3): replay each chunk with carried-in state, emit
// y = sum_n h_n * C_n + u * D. B and C rows staged in LDS (16 KB).
// ---------------------------------------------------------------------------
__global__ void k_scan_phase3(const float* __restrict__ u, const float* __restrict__ xdbl,
                              const float* __restrict__ delta, const float* __restrict__ logA,
                              const float* __restrict__ Dp, const float* __restrict__ carry,
                              float* __restrict__ y) {
  const int dblk = blockIdx.x & 3;
  const int ch   = (blockIdx.x >> 2) & (NCH - 1);
  const int b    = blockIdx.x >> 6;
  const int d    = dblk * 256 + threadIdx.x;
  const int t0   = ch * LCH;

  __shared__ float Bs[LCH][DSTATE];
  __shared__ float Cs[LCH][DSTATE];
  for (int i = threadIdx.x; i < LCH * DSTATE; i += 256) {
    int t = i >> 4, n = i & 15;
    size_t rowoff = (size_t)(b * LENGTH + t0 + t) * XPROJ_N;
    Bs[t][n] = xdbl[rowoff + DTRANK + n];
    Cs[t][n] = xdbl[rowoff + DTRANK + DSTATE + n];
  }
  __syncthreads();

  float An[DSTATE], h[DSTATE];
  size_t cbase = (((size_t)(b * DINNER + d)) * NCH + ch) * DSTATE;
#pragma unroll
  for (int n = 0; n < DSTATE; ++n) {
    An[n] = -__expf(logA[d * DSTATE + n]);
    h[n]  = carry[cbase + n];
  }
  const float Dd = Dp[d];
  for (int t = 0; t < LCH; ++t) {
    size_t idx = (size_t)(b * LENGTH + t0 + t) * DINNER + d;
    float dlt = delta[idx];
    float uu  = u[idx];
    float du  = dlt * uu;
    float acc = 0.f;
#pragma unroll
    for (int n = 0; n < DSTATE; ++n) {
      float a = __expf(dlt * An[n]);
      h[n] = a * h[n] + du * Bs[t][n];
      acc += h[n] * Cs[t][n];
    }
    y[idx] = acc + uu * Dd;                        // coalesced in d
  }
}

// ---------------------------------------------------------------------------
extern "C" void kernel_launch(void* const* d_in, const int* in_sizes, int n_in,
                              void* d_out, int out_size, void* d_ws, size_t ws_size,
                              hipStream_t stream) {
  (void)in_sizes; (void)n_in; (void)out_size; (void)ws_size;
  const float* u    = (const float*)d_in[0];
  const float* Wx   = (const float*)d_in[1];
  const float* Wdt  = (const float*)d_in[2];
  const float* bdt  = (const float*)d_in[3];
  const float* logA = (const float*)d_in[4];
  const float* Dp   = (const float*)d_in[5];
  float* y = (float*)d_out;

  // workspace layout (floats)
  float* ws     = (float*)d_ws;
  float* xdbl   = ws;                                   // 4096*96
  float* delta  = xdbl  + (size_t)MROWS * XPROJ_N;      // 4096*1024
  float* chunkA = delta + (size_t)MROWS * DINNER;       // 2*1024*16*16
  float* chunkH = chunkA + (size_t)BATCH * DINNER * NCH * DSTATE;
  float* carry  = chunkH + (size_t)BATCH * DINNER * NCH * DSTATE;

  // GEMM1: 256 x 6 tiles, 8 waves/block
  {
    int tiles = (MROWS / 16) * (XPROJ_N / 16);          // 1536
    k_xproj<<<(tiles + 7) / 8, 256, 0, stream>>>(u, Wx, xdbl);
  }
  // GEMM2: 256 x 64 tiles
  {
    int tiles = (MROWS / 16) * (DINNER / 16);           // 16384
    k_dtproj<<<(tiles + 7) / 8, 256, 0, stream>>>(xdbl, Wdt, bdt, delta);
  }
  // chunked scan
  const int scan_blocks = BATCH * NCH * (DINNER / 256); // 128
  k_scan_phase1<<<scan_blocks, 256, 0, stream>>>(u, xdbl, delta, logA, chunkA, chunkH);
  k_scan_phase2<<<(BATCH * DINNER * DSTATE) / 256, 256, 0, stream>>>(chunkA, chunkH, carry);
  k_scan_phase3<<<scan_blocks, 256, 0, stream>>>(u, xdbl, delta, logA, Dp, carry, y);
}